// RelPosEncoder_7567732376211
// MI455X (gfx1250) — compile-verified
//
#include <hip/hip_runtime.h>
#include <stdint.h>

// RelPosEncoder: out[i,j,:] = W[p,:] + T[t,:]  (one-hot gather; T folds e*W65 + W66+q + b)
// B=1, N=1024, C_Z=128, NO_BINS=72. Output 512MB f32 -> pure store-bandwidth bound.
// CDNA5 paths used: global_load_async_to_lds_b128 staging (+ s_wait_asynccnt),
// non-temporal b128 stores for the 512MB (>192MB L2) output stream.

namespace {
constexpr int kN    = 1024;
constexpr int kC    = 128;
constexpr int kBins = 72;
}

typedef float v4f __attribute__((ext_vector_type(4)));
typedef int   v4i __attribute__((vector_size(16)));   // matches async builtin param type

#if defined(__gfx1250__) && __has_builtin(__builtin_amdgcn_global_load_async_to_lds_b128)
#define USE_ASYNC_LDS 1
typedef __attribute__((address_space(1))) v4i as1_v4i;
typedef __attribute__((address_space(3))) v4i as3_v4i;
#define ASYNC_CP16(gsrc, ldst)                                                   \
  __builtin_amdgcn_global_load_async_to_lds_b128((as1_v4i*)(gsrc),               \
                                                 (as3_v4i*)(ldst), 0, 0)
#else
#define USE_ASYNC_LDS 0
#endif

__global__ __launch_bounds__(256, 1) void relpos_gather_kernel(
    const int* __restrict__ residue_index,
    const int* __restrict__ asym_id,
    const int* __restrict__ sym_id,
    const int* __restrict__ entity_id,
    const float* __restrict__ W,
    const float* __restrict__ bias,
    float* __restrict__ out)
{
  __shared__ float sW[kBins * kC];   // 36 KB raw W (rows 0..64 used in hot loop)
  __shared__ float sT[6 * kC];       // 3 KB: T[t] = (t<5 ? W65 : 0) + W[66+t] + b
  __shared__ int4  sIDs[kN];         // 16 KB packed {residue, asym, sym, entity}
  __shared__ int   sR[kN];
  __shared__ int   sA[kN];
  __shared__ int   sS[kN];
  __shared__ int   sE[kN];

  const int tid = threadIdx.y * 32 + threadIdx.x;  // 0..255

#if USE_ASYNC_LDS
  // CDNA5 async global->LDS staging (tracked by ASYNCcnt, no VGPR round-trip).
  for (int q = tid; q < (kBins * kC) / 4; q += 256) {   // 2304 x b128
    ASYNC_CP16(W + 4 * q, sW + 4 * q);
  }
  {
    const int q = tid;  // kN/4 == 256 == blockDim: exactly one b128 per thread/array
    ASYNC_CP16(residue_index + 4 * q, sR + 4 * q);
    ASYNC_CP16(asym_id       + 4 * q, sA + 4 * q);
    ASYNC_CP16(sym_id        + 4 * q, sS + 4 * q);
    ASYNC_CP16(entity_id     + 4 * q, sE + 4 * q);
  }
#if __has_builtin(__builtin_amdgcn_s_wait_asynccnt)
  __builtin_amdgcn_s_wait_asynccnt(0);
#else
  asm volatile("s_wait_asynccnt 0" ::: "memory");
#endif
#else
  for (int idx = tid; idx < kBins * kC; idx += 256) sW[idx] = W[idx];
  for (int idx = tid; idx < kN; idx += 256) {
    sR[idx] = residue_index[idx];
    sA[idx] = asym_id[idx];
    sS[idx] = sym_id[idx];
    sE[idx] = entity_id[idx];
  }
#endif
  __syncthreads();

  // Fold the 6 reachable (entity_same, rel_chain) combos + bias into sT:
  //   esame==1 -> t=q in 0..4 : W65 + W[66+t] + b
  //   esame==0 -> t=5         :       W[71]   + b
  for (int idx = tid; idx < 6 * kC; idx += 256) {
    const int t = idx >> 7, c = idx & (kC - 1);
    float v = sW[(66 + t) * kC + c] + bias[c];
    if (t < 5) v += sW[65 * kC + c];
    sT[idx] = v;
  }
  // Pack ids so the hot loop needs one broadcast ds_load_b128 per j.
  for (int idx = tid; idx < kN; idx += 256) {
    sIDs[idx] = make_int4(sR[idx], sA[idx], sS[idx], sE[idx]);
  }
  __syncthreads();

  const int i  = blockIdx.x;
  const int4 idi = sIDs[i];
  const int ri = idi.x, ai = idi.y, si = idi.z, ei = idi.w;
  const int lane = threadIdx.x;  // 0..31 -> channel quad (lane*4 .. lane*4+3)
  const int wv   = threadIdx.y;  // 0..7  -> j stripe

  const float* __restrict__ sWq = sW + 4 * lane;
  const float* __restrict__ sTq = sT + 4 * lane;
  float* const outRow = out + (size_t)i * kN * kC + 4 * lane;

  for (int j = wv; j < kN; j += 8) {
    // Wave-uniform per-j index math (j is uniform across the wave32).
    const int4 id = sIDs[j];                        // one broadcast ds_load_b128
    int p = min(max(ri - id.x + 32, 0), 63);        // clip(offset+32, 0, 2*32-1)
    p = (ai == id.y) ? p : 64;
    int t = (ei == id.w) ? min(max(si - id.z + 2, 0), 4) : 5;

    v4f acc = *(const v4f*)(sWq + p * kC) + *(const v4f*)(sTq + t * kC);

    // 512MB output stream (>> 192MB L2): non-temporal b128 store; each wave
    // writes one contiguous 512B output row.
    __builtin_nontemporal_store(acc, (v4f*)(outRow + (size_t)j * kC));
  }
}

extern "C" void kernel_launch(void* const* d_in, const int* in_sizes, int n_in,
                              void* d_out, int out_size, void* d_ws, size_t ws_size,
                              hipStream_t stream) {
  (void)in_sizes; (void)n_in; (void)out_size; (void)d_ws; (void)ws_size;
  const int*   residue_index = (const int*)d_in[0];
  const int*   asym_id       = (const int*)d_in[1];
  const int*   sym_id        = (const int*)d_in[2];
  const int*   entity_id     = (const int*)d_in[3];
  const float* W             = (const float*)d_in[4];
  const float* bias          = (const float*)d_in[5];
  float*       out           = (float*)d_out;

  dim3 block(32, 8, 1);   // 8 wave32s
  dim3 grid(kN, 1, 1);    // one block per i
  relpos_gather_kernel<<<grid, block, 0, stream>>>(
      residue_index, asym_id, sym_id, entity_id, W, bias, out);
}